// FeatureTokenizer_47682726921035
// MI455X (gfx1250) — compile-verified
//
#include <hip/hip_runtime.h>

#define NF 256
#define DT 512
#define LN_EPS 1e-5f
#define RS 516   // LDS row stride in dwords (512 + 4 pad -> halves land 32 banks apart)

typedef __attribute__((ext_vector_type(2))) float v2f;
typedef __attribute__((ext_vector_type(4))) float v4f;
typedef __attribute__((ext_vector_type(8))) float v8f;

// One block = 16 batch rows x one feature f x full d=512.
// 4 waves; wave w owns d slice [w*128, w*128+128) in both phases.
__global__ __launch_bounds__(128) void
FeatureTokenizer_47682726921035_kernel(const float* __restrict__ values,
                                       const float* __restrict__ masks,
                                       const float* __restrict__ W,     // (F, 2, D)
                                       const float* __restrict__ bias,  // (F, D)
                                       const float* __restrict__ gamma, // (D)
                                       const float* __restrict__ beta,  // (D)
                                       float* __restrict__ out)         // (B, F, D)
{
    const int f    = blockIdx.y;
    const int b0   = blockIdx.x << 4;
    const int tid  = threadIdx.x;
    const int wave = tid >> 5;             // 0..3 (wave32)
    const int lane = tid & 31;
    const int col  = lane & 15;            // N (d within tile) / M for A loads
    const bool lo  = lane < 16;
    const int dbase = wave << 7;           // wave's 128-wide d slice

    __shared__ __attribute__((aligned(16))) float s_tok[16 * RS]; // raw tokens
    __shared__ float s_sum[4][16];
    __shared__ float s_sq [4][16];
    __shared__ float s_mean[16];
    __shared__ float s_rstd[16];

    // ---- A matrix 16x4 f32. Lanes 0-15: K=0 (value), K=1 (mask).
    //      Lanes 16-31 (K=2,3): zero pad. Branch-free: load all, cndmask hi half.
    const int arow = b0 + col;
    const float ax = values[arow * NF + f];
    const float ay = masks [arow * NF + f];
    v2f a;
    a.x = lo ? ax : 0.0f;
    a.y = lo ? ay : 0.0f;

    float psum[8], psq[8];
#pragma unroll
    for (int r = 0; r < 8; ++r) { psum[r] = 0.0f; psq[r] = 0.0f; }

    // LDS scatter base (dword index): row = (lo ? r : r+8), d = dbase + t*16 + col
    const int lbase = (lo ? 0 : 8 * RS) + dbase + col;

#pragma unroll
    for (int t = 0; t < 8; ++t) {
        const int d = dbase + (t << 4) + col;

        // ---- B matrix 4x16 f32: K=0 -> W[f,0,d], K=1 -> W[f,1,d]; K=2,3 zero.
        const float w0 = W[(f * 2 + 0) * DT + d];
        const float w1 = W[(f * 2 + 1) * DT + d];
        v2f bm;
        bm.x = lo ? w0 : 0.0f;
        bm.y = lo ? w1 : 0.0f;

        // ---- C = bias broadcast down the column (bias depends only on d)
        const float bb = bias[f * DT + d];
        v8f c;
#pragma unroll
        for (int r = 0; r < 8; ++r) c[r] = bb;

        // D = A x B + C  ->  raw tokens tile (16 rows x 16 cols)
        const v8f acc = __builtin_amdgcn_wmma_f32_16x16x4_f32(
            /*neg_a=*/false, a, /*neg_b=*/false, bm,
            /*c_mod=*/(short)0, c, /*reuse_a=*/false, /*reuse_b=*/false);

#pragma unroll
        for (int r = 0; r < 8; ++r) {
            const float v = acc[r];
            psum[r] += v;
            psq[r]  += v * v;
            s_tok[lbase + r * RS + (t << 4)] = v;   // conflict-free scatter
        }
    }

    // ---- Row stats: VGPR r across a 16-lane half is one matrix row.
#pragma unroll
    for (int r = 0; r < 8; ++r) {
#pragma unroll
        for (int m = 1; m < 16; m <<= 1) {
            psum[r] += __shfl_xor(psum[r], m, 32);
            psq[r]  += __shfl_xor(psq[r],  m, 32);
        }
    }
    if (col == 0) { // lane 0 -> rows 0..7, lane 16 -> rows 8..15
        const int rbase = lo ? 0 : 8;
#pragma unroll
        for (int r = 0; r < 8; ++r) {
            s_sum[wave][rbase + r] = psum[r];
            s_sq [wave][rbase + r] = psq[r];
        }
    }
    __syncthreads();

    if (tid < 16) {
        const float s = s_sum[0][tid] + s_sum[1][tid] + s_sum[2][tid] + s_sum[3][tid];
        const float q = s_sq [0][tid] + s_sq [1][tid] + s_sq [2][tid] + s_sq [3][tid];
        const float mean = s * (1.0f / DT);
        const float var  = q * (1.0f / DT) - mean * mean;   // population var (ddof=0)
        s_mean[tid] = mean;
        s_rstd[tid] = rsqrtf(var + LN_EPS);
    }
    __syncthreads();

    // ---- Read-back + normalize + coalesced NT stores.
    //      Wave w re-reads its own d slice; each instruction stores 512B/wave.
    const int d0 = dbase + (lane << 2);                 // 4 floats per lane
    const v4f g4 = *(const v4f*)&gamma[d0];
    const v4f t4 = *(const v4f*)&beta [d0];
    const size_t outbase = ((size_t)b0 * NF + f) * DT + d0;

#pragma unroll
    for (int m = 0; m < 16; ++m) {
        const float mu = s_mean[m];
        const float rs = s_rstd[m];
        const v4f tok = *(const v4f*)&s_tok[m * RS + d0];
        const v4f o = (tok - mu) * rs * g4 + t4;
        __builtin_nontemporal_store(
            o, (v4f*)(out + outbase + (size_t)m * (NF * DT)));
    }
}

extern "C" void kernel_launch(void* const* d_in, const int* in_sizes, int n_in,
                              void* d_out, int out_size, void* d_ws, size_t ws_size,
                              hipStream_t stream) {
    const float* values = (const float*)d_in[0];
    const float* masks  = (const float*)d_in[1];
    const float* W      = (const float*)d_in[2];
    const float* bias   = (const float*)d_in[3];
    const float* gamma  = (const float*)d_in[4];
    const float* beta   = (const float*)d_in[5];
    float* out = (float*)d_out;

    const int batch = in_sizes[0] / NF;        // 2048
    dim3 grid(batch / 16, NF);                 // (128, 256)
    dim3 block(128);                           // 4 wave32s
    FeatureTokenizer_47682726921035_kernel<<<grid, block, 0, stream>>>(
        values, masks, W, bias, gamma, beta, out);
}